// Top2Router_4260607557959
// MI455X (gfx1250) — compile-verified
//
#include <hip/hip_runtime.h>

#define D_MODEL       1024
#define N_EXPERTS     8
#define TOKENS_PER_WG 128
#define KC            64                 // K-chunk staged in LDS (double buffered)
#define NCHUNK        (D_MODEL / KC)     // 16
#define XSTR          (KC + 4)           // 68 floats: padded row stride (bank-conflict free)
#define XBUF          (TOKENS_PER_WG * XSTR)   // 8704 floats per buffer
#define WSTR          (D_MODEL + 4)      // 1028 floats
#define LSTR          9                  // logits row stride

typedef __attribute__((ext_vector_type(2))) float v2f;
typedef __attribute__((ext_vector_type(8))) float v8f;

__global__ __launch_bounds__(256)
void top2_router_kernel(const float* __restrict__ x,
                        const float* __restrict__ gate_w,
                        const float* __restrict__ gate_b,
                        float* __restrict__ out,
                        int n_tokens) {
    __shared__ __align__(16) float xtile[2 * XBUF];   // 69632 B, double-buffered x chunks
    __shared__ __align__(16) float wtile[16 * WSTR];  // 65792 B (rows 8..15 zero)

    const int tid  = threadIdx.x;
    const int lane = tid & 31;
    const int wv   = tid >> 5;                    // wave id 0..7
    const int tok0 = blockIdx.x * TOKENS_PER_WG;

    // ---- Stage gate_w into LDS (rows 0..7 real, 8..15 zero) ----
    #pragma unroll
    for (int i = 0; i < 8; ++i) {                 // 2048 float4 total / 256 threads
        int q  = tid + i * 256;
        int e  = q >> 8;                          // 256 float4 per expert row
        int c4 = q & 255;
        float4 v = reinterpret_cast<const float4*>(gate_w + e * D_MODEL)[c4];
        *reinterpret_cast<float4*>(&wtile[e * WSTR + c4 * 4]) = v;
    }
    for (int i = tid; i < 8 * WSTR; i += 256)
        wtile[8 * WSTR + i] = 0.0f;

    // LDS byte address of xtile base (flat shared pointer: low 32 bits = LDS offset)
    const uint32_t xlds0 = (uint32_t)(uintptr_t)&xtile[0];

    // Issue one chunk's async global->LDS copies: 128 rows x 64 floats = 2048 b128,
    // 8 async instructions per thread (per-lane 16B, fully coalesced).
    auto issue_chunk = [&](int c, int buf) {
        const uint32_t dst0 = xlds0 + (uint32_t)buf * (XBUF * 4u);
        const uint32_t kc0b = (uint32_t)(c * KC) * 4u;     // byte offset within a row
        #pragma unroll
        for (int i = 0; i < 8; ++i) {
            int q   = tid + i * 256;
            int row = q >> 4;                              // 16 b128 per token row
            int c4  = q & 15;
            uint32_t voff = (uint32_t)(tok0 + row) * (D_MODEL * 4u) + kc0b + (uint32_t)c4 * 16u;
            uint32_t dlds = dst0 + (uint32_t)(row * XSTR + c4 * 4) * 4u;
            asm volatile("global_load_async_to_lds_b128 %0, %1, %2"
                         :: "v"(dlds), "v"(voff), "s"(x)
                         : "memory");
        }
    };

    // ---- WMMA fragment addressing (ISA 16x16x4 f32 layouts) ----
    const int rowA = lane & 15;                   // A: M = lane&15 in both half-waves
    const int nCol = lane & 15;                   // B: N = lane&15
    const int kk   = (lane >> 4) << 1;            // K sub-offset: 0 (lanes 0-15) or 2 (lanes 16-31)
    const float* brow = &wtile[nCol * WSTR];

    v8f acc = {0.f, 0.f, 0.f, 0.f, 0.f, 0.f, 0.f, 0.f};

    issue_chunk(0, 0);
    for (int c = 0; c < NCHUNK; ++c) {
        if (c + 1 < NCHUNK) {
            issue_chunk(c + 1, (c + 1) & 1);      // prefetch next chunk into other buffer
            asm volatile("s_wait_asynccnt 0x8" ::: "memory");   // drain chunk c (in-order)
        } else {
            asm volatile("s_wait_asynccnt 0x0" ::: "memory");
        }
        __syncthreads();                          // chunk c visible to all waves

        const float* arow = &xtile[(c & 1) * XBUF + (wv * 16 + rowA) * XSTR];
        const int kc0 = c * KC;
        #pragma unroll
        for (int k = 0; k < KC; k += 4) {
            v2f a = *reinterpret_cast<const v2f*>(arow + k + kk);
            v2f b = *reinterpret_cast<const v2f*>(brow + kc0 + k + kk);
            acc = __builtin_amdgcn_wmma_f32_16x16x4_f32(
                      false, a, false, b, (short)0, acc, false, false);
        }
        __syncthreads();                          // all waves done reading buffer c&1
    }

    // ---- scatter logits to LDS (reuse xtile) ----
    float* lgts = xtile;                          // [128][LSTR]
    {
        const int m0 = wv * 16 + ((lane >> 4) << 3);  // C/D: VGPR r -> rows r (lo), r+8 (hi)
        if (nCol < N_EXPERTS) {
            #pragma unroll
            for (int r = 0; r < 8; ++r)
                lgts[(m0 + r) * LSTR + nCol] = acc[r];
        }
    }
    __syncthreads();

    // ---- softmax + top-2 per token (threads 0..127) ----
    if (tid < TOKENS_PER_WG) {
        const int t = tid;
        const size_t gtok = (size_t)tok0 + t;

        float lg[N_EXPERTS];
        float mx = -3.402823466e+38f;
        #pragma unroll
        for (int e = 0; e < N_EXPERTS; ++e) {
            lg[e] = lgts[t * LSTR + e] + gate_b[e];
            mx = fmaxf(mx, lg[e]);
        }
        float p[N_EXPERTS];
        float s = 0.f;
        #pragma unroll
        for (int e = 0; e < N_EXPERTS; ++e) { p[e] = __expf(lg[e] - mx); s += p[e]; }
        const float inv = 1.0f / s;

        // top-2, ties resolve to lowest index (strict >)
        int i0 = 0; float v0 = p[0];
        #pragma unroll
        for (int e = 1; e < N_EXPERTS; ++e) if (p[e] > v0) { v0 = p[e]; i0 = e; }
        int i1 = (i0 == 0) ? 1 : 0; float v1 = p[i1];
        #pragma unroll
        for (int e = 0; e < N_EXPERTS; ++e)
            if (e != i0 && p[e] > v1) { v1 = p[e]; i1 = e; }

        float* vals = out;                               // [T,2] values
        float* idxs = out + (size_t)2 * n_tokens;        // [T,2] indices (as float)
        vals[gtok * 2 + 0] = v0 * inv;
        vals[gtok * 2 + 1] = v1 * inv;
        idxs[gtok * 2 + 0] = (float)i0;
        idxs[gtok * 2 + 1] = (float)i1;
    }
}

extern "C" void kernel_launch(void* const* d_in, const int* in_sizes, int n_in,
                              void* d_out, int out_size, void* d_ws, size_t ws_size,
                              hipStream_t stream) {
    const float* x  = (const float*)d_in[0];   // [8, 8192, 1024]
    const float* gw = (const float*)d_in[1];   // [8, 1024]
    const float* gb = (const float*)d_in[2];   // [8]
    float* out = (float*)d_out;                // values [T,2] ++ indices [T,2]

    const int n_tokens = in_sizes[0] / D_MODEL;          // 65536
    const int grid = n_tokens / TOKENS_PER_WG;           // 512

    hipLaunchKernelGGL(top2_router_kernel, dim3(grid), dim3(256), 0, stream,
                       x, gw, gb, out, n_tokens);
}